// EatRxnLayer_16277926052098
// MI455X (gfx1250) — compile-verified
//
#include <hip/hip_runtime.h>
#include <cstdint>

// EatRxn layer, MI455X (gfx1250).
// Output is 258 MiB, mostly zeros -> pure store-bound (~11.5 us at 23.3 TB/s).
// Strategy: single pass, one block per batch element. Non-zero entries
// (rows h,p / cols h,p / corners) are computed and staged in LDS; the zero
// bulk is emitted with gfx1250 async LDS->global b128 stores (ASYNCcnt path),
// reading from a zeroed LDS buffer. No WMMA: there is no matrix product here.

#define NTOT  128
#define NBATCH 4096

__global__ __launch_bounds__(256)
void eat_rxn_kernel(const float* __restrict__ mu,
                    const float* __restrict__ ncov,
                    float* __restrict__ out)
{
    const int b   = blockIdx.x;
    const int tid = threadIdx.x;

    float* outMu  = out;                              // NBATCH*NTOT
    float* outCov = out + (size_t)NBATCH * NTOT;      // NBATCH*NTOT*NTOT
    const float* covb = ncov + (size_t)b * NTOT * NTOT;
    float* ocov       = outCov + (size_t)b * NTOT * NTOT;

    __shared__ float row0[NTOT];      // ncovTE[b,0,:]
    __shared__ float row1[NTOT];      // ncovTE[b,1,:]
    __shared__ float colv[2 * NTOT];  // per row j>=2: ( n3_j, -n3_j )
    __shared__ float zbuf[1024];      // 4 KB of zeros feeding async stores

    // Zero the async-store source buffer (16 B per lane of a 256-thread block).
    zbuf[tid]       = 0.0f;
    zbuf[tid + 256] = 0.0f;
    zbuf[tid + 512] = 0.0f;
    zbuf[tid + 768] = 0.0f;

    if (tid < NTOT) {
        const int j = tid;
        // Uniform scalars (compiler scalarizes these to s_load).
        const float mh  = mu[(size_t)b * NTOT + 0];
        const float mp  = mu[(size_t)b * NTOT + 1];
        const float chp = covb[0 * NTOT + 1];  // ncov[h,p]
        const float cph = covb[1 * NTOT + 0];  // ncov[p,h]  (input not symmetric)
        const float chh = covb[0 * NTOT + 0];
        const float cpp = covb[1 * NTOT + 1];

        // Per-lane data: mu[b,j] and the two covariance columns.
        const float muj = mu[(size_t)b * NTOT + j];
        const float cjh = covb[(size_t)j * NTOT + 0];
        const float cjp = covb[(size_t)j * NTOT + 1];

        // n3_j, evaluated in the reference's term order.
        float n3 = -2.0f * muj * (mp * mh) + muj * cph + mp * cjh + mh * cjp;
        if (j < 2) n3 = 0.0f;  // masked for j==h, j==p

        // Third moments for the 2x2 corner.
        const float n3_pph = -2.0f * mp * mp * mh + mp * cph + mp * cph + mh * cpp;
        const float n3_hhp = -2.0f * mh * mh * mp + mh * chp + mh * chp + mp * chh;
        const float n3_hpp = -2.0f * mh * mp * mp + mh * cpp + mp * chp + mp * chp;
        const float vals_pp = -2.0f * n3_pph + chp;
        const float vals_hh =  2.0f * n3_hhp + chp;
        const float vals_hp = -n3_hhp + n3_hpp - chp;

        row0[j] = (j == 0) ? vals_hh : (j == 1) ? vals_hp :  n3;
        row1[j] = (j == 0) ? vals_hp : (j == 1) ? vals_pp : -n3;
        colv[2 * j + 0] =  n3;   // ncovTE[b,j,0], j>=2
        colv[2 * j + 1] = -n3;   // ncovTE[b,j,1], j>=2

        // muTE row (covers all 128 entries; output is poisoned).
        outMu[(size_t)b * NTOT + j] = (j == 0) ? chp : (j == 1) ? -chp : 0.0f;
    }
    __syncthreads();

    // LDS byte address of this lane's 16 B zero chunk (low 32 bits of the
    // generic pointer == LDS offset on gfx1250).
    const uint32_t zoff =
        (uint32_t)(uintptr_t)(&zbuf[0]) + (uint32_t)(tid & 31) * 16u;
    const uint64_t sbase = (uint64_t)(uintptr_t)outCov;
    const uint32_t bbase = (uint32_t)b * (uint32_t)(NTOT * NTOT * 4);  // < 2^28

    // 128x128 floats = 4096 chunks of 16 B; 256 threads x 16 iterations.
    #pragma unroll 1
    for (int it = 0; it < 16; ++it) {
        const int c = tid + it * 256;   // chunk id, row r = c/32, quad q = c%32
        const int r = c >> 5;
        const int q = c & 31;
        if (r < 2) {
            const float* src = (r == 0) ? row0 : row1;
            float4 v = make_float4(src[q * 4 + 0], src[q * 4 + 1],
                                   src[q * 4 + 2], src[q * 4 + 3]);
            *reinterpret_cast<float4*>(ocov + (size_t)c * 4) = v;
        } else if (q == 0) {
            float4 v = make_float4(colv[2 * r], colv[2 * r + 1], 0.0f, 0.0f);
            *reinterpret_cast<float4*>(ocov + (size_t)c * 4) = v;
        } else {
            // Bulk zeros: gfx1250 async LDS->global store, 16 B/lane.
            const uint32_t voff = bbase + (uint32_t)c * 16u;
            asm volatile("global_store_async_from_lds_b128 %0, %1, %2"
                         :
                         : "v"(voff), "v"(zoff), "s"(sbase)
                         : "memory");
        }
    }

    // Drain outstanding async stores (s_endpgm would also wait-idle).
#if defined(__has_builtin) && __has_builtin(__builtin_amdgcn_s_wait_asynccnt)
    __builtin_amdgcn_s_wait_asynccnt(0);
#else
    asm volatile("s_wait_asynccnt 0x0" ::: "memory");
#endif
}

extern "C" void kernel_launch(void* const* d_in, const int* in_sizes, int n_in,
                              void* d_out, int out_size, void* d_ws, size_t ws_size,
                              hipStream_t stream) {
    (void)in_sizes; (void)n_in; (void)out_size; (void)d_ws; (void)ws_size;
    const float* mu   = (const float*)d_in[0];
    const float* ncov = (const float*)d_in[1];
    float* out = (float*)d_out;
    eat_rxn_kernel<<<dim3(NBATCH), dim3(256), 0, stream>>>(mu, ncov, out);
}